// PosNet_12884901888473
// MI455X (gfx1250) — compile-verified
//
#include <hip/hip_runtime.h>
#include <hip/hip_bf16.h>

// ---------------- problem constants (match reference) ----------------
#define GN      50000
#define GE      800000
#define BN_EPS  1e-5f
#define LSLOPE  0.01f
#define NLAYERS 13
#define HMAX    512

static const int Hdims[15] = {16,16,32,64,128,256,256,512,256,256,128,64,32,16,3};

typedef float v2f __attribute__((ext_vector_type(2)));
typedef float v8f __attribute__((ext_vector_type(8)));

#if __has_builtin(__builtin_amdgcn_wmma_f32_16x16x4_f32)
#define HAVE_WMMA_F32 1
#endif

// ---------------------------------------------------------------------
// 1) degree count over dst (self-loop added analytically later)
__global__ void deg_count_kernel(const int* __restrict__ dst, int* __restrict__ deg) {
    int e = blockIdx.x * blockDim.x + threadIdx.x;
    if (e < GE) atomicAdd(&deg[dst[e]], 1);
}

// 2) dis[i] = rsqrt(deg_i + 1)   (deg includes self loop => always >= 1)
__global__ void dis_kernel(const int* __restrict__ deg, float* __restrict__ dis) {
    int i = blockIdx.x * blockDim.x + threadIdx.x;
    if (i < GN) dis[i] = rsqrtf((float)(deg[i] + 1));
}

// 3) per-edge norm = dis[src]*dis[dst]
__global__ void enorm_kernel(const int* __restrict__ src, const int* __restrict__ dst,
                             const float* __restrict__ dis, float* __restrict__ enorm) {
    int e = blockIdx.x * blockDim.x + threadIdx.x;
    if (e < GE) enorm[e] = dis[src[e]] * dis[dst[e]];
}

// 4) column-wise min/max of z1 (N x 16), single block reduction.
//    zstat[0..15] = (min+max)/2 per col ; zstat[16] = 1 / max_col(max-min)
__global__ void zstat_kernel(const float* __restrict__ z, float* __restrict__ zstat) {
    __shared__ float smin[256][16];
    __shared__ float smax[256][16];
    const int tid = threadIdx.x;
    float mn[16], mx[16];
#pragma unroll
    for (int c = 0; c < 16; ++c) { mn[c] = 3.4e38f; mx[c] = -3.4e38f; }
    for (int i = tid; i < GN; i += 256) {
        const float* row = z + (size_t)i * 16;
#pragma unroll
        for (int c = 0; c < 16; ++c) {
            float v = row[c];
            mn[c] = fminf(mn[c], v);
            mx[c] = fmaxf(mx[c], v);
        }
    }
#pragma unroll
    for (int c = 0; c < 16; ++c) { smin[tid][c] = mn[c]; smax[tid][c] = mx[c]; }
    __syncthreads();
    for (int off = 128; off > 0; off >>= 1) {
        if (tid < off) {
#pragma unroll
            for (int c = 0; c < 16; ++c) {
                smin[tid][c] = fminf(smin[tid][c], smin[tid + off][c]);
                smax[tid][c] = fmaxf(smax[tid][c], smax[tid + off][c]);
            }
        }
        __syncthreads();
    }
    if (tid == 0) {
        float sc = 0.f;
#pragma unroll
        for (int c = 0; c < 16; ++c) {
            sc = fmaxf(sc, smax[0][c] - smin[0][c]);
            zstat[c] = 0.5f * (smin[0][c] + smax[0][c]);
        }
        zstat[16] = 1.0f / sc;
    }
}

// 5) X = (z1 - zc) * inv_scale
__global__ void rescale_kernel(const float* __restrict__ z, const float* __restrict__ zstat,
                               float* __restrict__ X) {
    int t = blockIdx.x * blockDim.x + threadIdx.x;
    if (t >= GN * 16) return;
    int c = t & 15;
    X[t] = (z[t] - zstat[c]) * zstat[16];
}

// ---------------------------------------------------------------------
// 6) Dense GEMM  C[N,Kout] = A[N,K] * B[K,Kout] with V_WMMA_F32_16X16X4_F32.
//    One wave owns one 16x16 output tile; K stepped by 4.
//    A frag (16x4):  lane<16 -> {K+0,K+1}, lane>=16 -> {K+2,K+3}, M = lane&15
//    B frag (4x16):  v0: rows K+0 / K+2, v1: rows K+1 / K+3, N = lane&15
//    C/D (16x16):    vgpr r: lanes0-15 M=r, lanes16-31 M=r+8, N = lane&15
__global__ void gemm_wmma_f32(const float* __restrict__ A, const float* __restrict__ B,
                              float* __restrict__ C, int K, int Kout,
                              int rowTiles, int colTiles) {
    int wave = blockIdx.x * (blockDim.x >> 5) + (threadIdx.x >> 5);
    if (wave >= rowTiles * colTiles) return;
    const int rt = wave / colTiles, ct = wave % colTiles;
    const int r0 = rt * 16, c0 = ct * 16;
    const int lane = threadIdx.x & 31;
    const int m = lane & 15, half = lane >> 4;
#ifdef HAVE_WMMA_F32
    v8f acc = {};
    const float* arow = A + (size_t)(r0 + m) * K + 2 * half;   // A[r0+m][k + 2*half]
    for (int k = 0; k < K; k += 4) {
        v2f a, b;
        a.x = arow[k];
        a.y = arow[k + 1];
        const int kb = k + 2 * half;
        b.x = B[(size_t)kb * Kout + c0 + m];
        b.y = B[(size_t)(kb + 1) * Kout + c0 + m];
        acc = __builtin_amdgcn_wmma_f32_16x16x4_f32(false, a, false, b,
                                                    (short)0, acc, false, false);
    }
    float* crow = C + (size_t)(r0 + 8 * half) * Kout + c0 + m;
#pragma unroll
    for (int r = 0; r < 8; ++r) crow[(size_t)r * Kout] = acc[r];
#else
    // scalar fallback with identical output mapping
    float acc[8];
#pragma unroll
    for (int r = 0; r < 8; ++r) acc[r] = 0.f;
    for (int k = 0; k < K; ++k) {
        float bv = B[(size_t)k * Kout + c0 + m];
#pragma unroll
        for (int r = 0; r < 8; ++r)
            acc[r] += A[(size_t)(r0 + 8 * half + r) * K + k] * bv;
    }
    float* crow = C + (size_t)(r0 + 8 * half) * Kout + c0 + m;
#pragma unroll
    for (int r = 0; r < 8; ++r) crow[(size_t)r * Kout] = acc[r];
#endif
}

// ---------------------------------------------------------------------
// 7) Edge scatter: AGG[dst] += norm_e * XW[src]   (float4 per thread)
__global__ void scatter_kernel(const float4* __restrict__ XW, const float* __restrict__ enorm,
                               const int* __restrict__ src, const int* __restrict__ dst,
                               float* __restrict__ AGG, int Kq, int Kout, long total) {
    long t = (long)blockIdx.x * blockDim.x + threadIdx.x;
    if (t >= total) return;
    int e  = (int)(t / Kq);
    int cq = (int)(t % Kq);
    float nrm = enorm[e];
    float4 v = XW[(size_t)src[e] * Kq + cq];
    float* out = AGG + (size_t)dst[e] * Kout + cq * 4;
    atomicAdd(out + 0, nrm * v.x);
    atomicAdd(out + 1, nrm * v.y);
    atomicAdd(out + 2, nrm * v.z);
    atomicAdd(out + 3, nrm * v.w);
}

// 8) out = agg + dis^2*xw + b  (in-place into AGG) + per-channel sum/sumsq
//    block = 16 channels x 16 row-lanes; grid = (Kout/16, RSLICE)
__global__ void stats_kernel(float* __restrict__ AGG, const float* __restrict__ XW,
                             const float* __restrict__ dis, const float* __restrict__ bias,
                             float* __restrict__ sum, float* __restrict__ sumsq, int Kout) {
    const int cl = threadIdx.x & 15;
    const int rl = threadIdx.x >> 4;
    const int ch = blockIdx.x * 16 + cl;
    const float b = bias[ch];
    float s = 0.f, ss = 0.f;
    for (int r = blockIdx.y * 16 + rl; r < GN; r += 16 * gridDim.y) {
        size_t idx = (size_t)r * Kout + ch;
        float d = dis[r];
        float v = AGG[idx] + d * d * XW[idx] + b;
        AGG[idx] = v;
        s += v; ss += v * v;
    }
    __shared__ float sh[2][16][16];
    sh[0][rl][cl] = s; sh[1][rl][cl] = ss;
    __syncthreads();
    for (int off = 8; off > 0; off >>= 1) {
        if (rl < off) {
            sh[0][rl][cl] += sh[0][rl + off][cl];
            sh[1][rl][cl] += sh[1][rl + off][cl];
        }
        __syncthreads();
    }
    if (rl == 0) {
        atomicAdd(&sum[ch],   sh[0][0][cl]);
        atomicAdd(&sumsq[ch], sh[1][0][cl]);
    }
}

// 9) per-channel BN affine params
__global__ void bnparam_kernel(const float* __restrict__ sum, const float* __restrict__ sumsq,
                               const float* __restrict__ gamma, const float* __restrict__ beta,
                               float* __restrict__ scale, float* __restrict__ shift, int Kout) {
    int c = blockIdx.x * blockDim.x + threadIdx.x;
    if (c >= Kout) return;
    const float invN = 1.0f / (float)GN;
    float mean = sum[c] * invN;
    float var  = sumsq[c] * invN - mean * mean;
    float inv  = rsqrtf(var + BN_EPS);
    float sc   = gamma[c] * inv;
    scale[c] = sc;
    shift[c] = beta[c] - mean * sc;
}

// 10) X = LeakyReLU(AGG*scale + shift)   (float4 per thread)
__global__ void norm_lrelu_kernel(const float4* __restrict__ AGG, const float* __restrict__ scale,
                                  const float* __restrict__ shift, float4* __restrict__ X,
                                  int Kq, long total) {
    long t = (long)blockIdx.x * blockDim.x + threadIdx.x;
    if (t >= total) return;
    int c = (int)(t % Kq) * 4;
    float4 v = AGG[t];
    float4 r;
    float y;
    y = v.x * scale[c + 0] + shift[c + 0]; r.x = y > 0.f ? y : LSLOPE * y;
    y = v.y * scale[c + 1] + shift[c + 1]; r.y = y > 0.f ? y : LSLOPE * y;
    y = v.z * scale[c + 2] + shift[c + 2]; r.z = y > 0.f ? y : LSLOPE * y;
    y = v.w * scale[c + 3] + shift[c + 3]; r.w = y > 0.f ? y : LSLOPE * y;
    X[t] = r;
}

// 11) out = x_pos + X(16) @ lin_w(16x3) + lin_b
__global__ void final_kernel(const float* __restrict__ X, const float* __restrict__ x_pos,
                             const float* __restrict__ lin_w, const float* __restrict__ lin_b,
                             float* __restrict__ out) {
    int i = blockIdx.x * blockDim.x + threadIdx.x;
    if (i >= GN) return;
    float acc0 = lin_b[0], acc1 = lin_b[1], acc2 = lin_b[2];
    const float* row = X + (size_t)i * 16;
#pragma unroll
    for (int c = 0; c < 16; ++c) {
        float xv = row[c];
        acc0 += xv * lin_w[c * 3 + 0];
        acc1 += xv * lin_w[c * 3 + 1];
        acc2 += xv * lin_w[c * 3 + 2];
    }
    out[i * 3 + 0] = x_pos[i * 3 + 0] + acc0;
    out[i * 3 + 1] = x_pos[i * 3 + 1] + acc1;
    out[i * 3 + 2] = x_pos[i * 3 + 2] + acc2;
}

// ---------------------------------------------------------------------
static inline long cdiv(long a, long b) { return (a + b - 1) / b; }

extern "C" void kernel_launch(void* const* d_in, const int* in_sizes, int n_in,
                              void* d_out, int out_size, void* d_ws, size_t ws_size,
                              hipStream_t stream) {
    // ---- inputs (dict order, lists flattened) ----
    const float* z1    = (const float*)d_in[0];
    const float* x_pos = (const float*)d_in[1];
    const int*   ei    = (const int*)d_in[2];
    const int*   src   = ei;        // edge_index[0]
    const int*   dst   = ei + GE;   // edge_index[1]
    const float *Ws[NLAYERS], *bs[NLAYERS], *gms[NLAYERS], *bts[NLAYERS];
    for (int i = 0; i < NLAYERS; ++i) {
        Ws[i]  = (const float*)d_in[3 + i];
        bs[i]  = (const float*)d_in[3 + NLAYERS + i];
        gms[i] = (const float*)d_in[3 + 2 * NLAYERS + i];
        bts[i] = (const float*)d_in[3 + 3 * NLAYERS + i];
    }
    const float* lin_w = (const float*)d_in[3 + 4 * NLAYERS];
    const float* lin_b = (const float*)d_in[3 + 4 * NLAYERS + 1];
    float* out = (float*)d_out;

    // ---- workspace layout ----
    char* p = (char*)d_ws;
    float* X     = (float*)p; p += (size_t)GN * HMAX * 4;
    float* XW    = (float*)p; p += (size_t)GN * HMAX * 4;
    float* AGG   = (float*)p; p += (size_t)GN * HMAX * 4;
    float* DIS   = (float*)p; p += (size_t)GN * 4;
    int*   DEG   = (int*)  p; p += (size_t)GN * 4;
    float* ENORM = (float*)p; p += (size_t)GE * 4;
    float* SUM   = (float*)p; p += HMAX * 4;
    float* SUMSQ = (float*)p; p += HMAX * 4;
    float* SCALE = (float*)p; p += HMAX * 4;
    float* SHIFT = (float*)p; p += HMAX * 4;
    float* ZSTAT = (float*)p; p += 32 * 4;

    const int B = 256;

    // ---- fixed-topology precompute ----
    hipMemsetAsync(DEG, 0, (size_t)GN * 4, stream);
    deg_count_kernel<<<cdiv(GE, B), B, 0, stream>>>(dst, DEG);
    dis_kernel<<<cdiv(GN, B), B, 0, stream>>>(DEG, DIS);
    enorm_kernel<<<cdiv(GE, B), B, 0, stream>>>(src, dst, DIS, ENORM);

    // ---- input rescale ----
    zstat_kernel<<<1, 256, 0, stream>>>(z1, ZSTAT);
    rescale_kernel<<<cdiv((long)GN * 16, B), B, 0, stream>>>(z1, ZSTAT, X);

    // ---- 13 GCN layers ----
    const int rowTiles = GN / 16;   // 3125
    for (int i = 0; i < NLAYERS; ++i) {
        const int K = Hdims[i], Kout = Hdims[i + 1];
        const int colTiles = Kout / 16;
        const int Kq = Kout / 4;

        // GEMM: XW = X * W    (v_wmma_f32_16x16x4_f32)
        long tiles = (long)rowTiles * colTiles;
        gemm_wmma_f32<<<cdiv(tiles, B / 32), B, 0, stream>>>(X, Ws[i], XW, K, Kout,
                                                             rowTiles, colTiles);

        hipMemsetAsync(AGG, 0, (size_t)GN * Kout * 4, stream);
        hipMemsetAsync(SUM, 0, (size_t)2 * HMAX * 4, stream);   // SUM+SUMSQ contiguous

        long etotal = (long)GE * Kq;
        scatter_kernel<<<cdiv(etotal, B), B, 0, stream>>>((const float4*)XW, ENORM, src, dst,
                                                          AGG, Kq, Kout, etotal);

        dim3 sgrid(Kout / 16, 64);
        stats_kernel<<<sgrid, 256, 0, stream>>>(AGG, XW, DIS, bs[i], SUM, SUMSQ, Kout);

        bnparam_kernel<<<cdiv(Kout, B), B, 0, stream>>>(SUM, SUMSQ, gms[i], bts[i],
                                                        SCALE, SHIFT, Kout);

        long ntotal = (long)GN * Kq;
        norm_lrelu_kernel<<<cdiv(ntotal, B), B, 0, stream>>>((const float4*)AGG, SCALE, SHIFT,
                                                             (float4*)X, Kq, ntotal);
    }

    // ---- final linear + residual ----
    final_kernel<<<cdiv(GN, B), B, 0, stream>>>(X, x_pos, lin_w, lin_b, out);
}